// DLEM_32495722561915
// MI455X (gfx1250) — compile-verified
//
#include <hip/hip_runtime.h>
#include <math.h>

// ---------------- problem constants (from reference) ----------------
#define BATCH   32
#define CIN     128
#define NSIG    8192
#define NH      8190        // conv1 output length (valid conv, k=3)
#define NOUT    8190        // output elements per batch
#define OC1     10
#define M16     16
#define K1      384         // 128 ic * 3 taps
#define OUT_TILE 124        // outputs per block
#define NTILES   67         // ceil(8190/124)
#define H_TILE   128        // h positions per block (OUT_TILE + 4 halo)
#define CHUNK_IC 32         // input channels staged per LDS pass
#define XS_POS   130        // H_TILE + 2 taps
#define XS_STR   132

typedef float v2f __attribute__((ext_vector_type(2)));
typedef float v8f __attribute__((ext_vector_type(8)));

// Fused pipeline: conv1(WMMA f32 16x16x4) -> relu -> conv2(1x1) -> relu ->
// convT(k=3) -> relu -> conv3(1x1) -> sigmoid -> mass stencil -> log
// Also emits one partial sum per block for the global-mean pass.
__global__ __launch_bounds__(256)
void dlem_pipeline(const float* __restrict__ signal,
                   const float* __restrict__ curr,
                   const float* __restrict__ w1, const float* __restrict__ b1,
                   const float* __restrict__ w2, const float* __restrict__ b2,
                   const float* __restrict__ wt, const float* __restrict__ bt,
                   const float* __restrict__ w3, const float* __restrict__ b3,
                   const float* __restrict__ cstp,
                   float* __restrict__ out,
                   float* __restrict__ partials)
{
    __shared__ float xs [CHUNK_IC * XS_STR];  // signal chunk: [ic][pos]
    __shared__ float wAs[M16 * K1];           // conv1 weights [m][k], rows 10..15 = 0
    __shared__ float h1s[M16 * H_TILE];       // conv1+relu   [ch][pos]
    __shared__ float h2s[OC1 * H_TILE];       // conv2+relu   [ch][pos]
    __shared__ float ys [H_TILE * OC1];       // convT+relu   [pos][ch]
    __shared__ float lefts[128], rights[128];
    __shared__ float b1s[16], b2s[16], w2s[100], wtA[304], bts[16], w3s[20], b3s[2];
    __shared__ float red[256];

    const int tid  = threadIdx.x;
    const int wv   = tid >> 5;          // wave id 0..7
    const int lane = tid & 31;
    const int half = lane >> 4;         // 0: lanes 0-15, 1: lanes 16-31
    const int col  = lane & 15;         // N column / M row within half

    const int b    = blockIdx.x / NTILES;
    const int tile = blockIdx.x % NTILES;
    const int o0   = tile * OUT_TILE;   // first output index of this block
    const int hb   = o0 - 2;            // first h position (with halo)
    const int sigb = b * (CIN * NSIG);

    // ---- stage small weights/biases ----
    for (int i = tid; i < M16 * K1; i += 256) {
        int m = i / K1, k = i - m * K1;
        wAs[i] = (m < OC1) ? w1[m * K1 + k] : 0.0f;
    }
    if (tid < 16) { b1s[tid] = (tid < OC1) ? b1[tid] : 0.0f; }
    if (tid < 16) { b2s[tid] = (tid < OC1) ? b2[tid] : 0.0f;
                    bts[tid] = (tid < OC1) ? bt[tid] : 0.0f; }
    for (int i = tid; i < 100; i += 256) w2s[i] = w2[i];
    for (int i = tid; i < 300; i += 256) {          // wtA[oc][ic][j] = wt[ic][oc][2-j]
        int oc = i / 30, r = i - oc * 30, ic = r / 3, j = r - ic * 3;
        wtA[i] = wt[(ic * 10 + oc) * 3 + (2 - j)];
    }
    if (tid < 20) w3s[tid] = w3[tid];
    if (tid < 2)  b3s[tid] = b3[tid];

    // ---- conv1: WMMA f32 16x16x4, K = 384 in 4 chunks of 96 ----
    v8f acc = {0.f, 0.f, 0.f, 0.f, 0.f, 0.f, 0.f, 0.f};
    const int nbase = wv * 16 + col;    // this lane's h-position column

    for (int cc = 0; cc < 4; ++cc) {
        __syncthreads();
        // stage 32 input channels x 130 positions (clamped addressing;
        // out-of-range h positions are zeroed after conv2)
        for (int i = tid; i < CHUNK_IC * XS_POS; i += 256) {
            int ic = i / XS_POS, q = i - ic * XS_POS;
            int g = hb + q;
            g = (g < 0) ? 0 : ((g > NSIG - 1) ? NSIG - 1 : g);
            xs[ic * XS_STR + q] = signal[sigb + (cc * CHUNK_IC + ic) * NSIG + g];
        }
        __syncthreads();

        #pragma unroll
        for (int ks = 0; ks < 24; ++ks) {
            const int kl = ks * 4 + half * 2;        // lane's K (chunk-local)
            const int kg = cc * 96 + kl;             // global K
            v2f a, bm;
            // A: 16x4 f32 — lanes 0-15: K,K+1 ; lanes 16-31: K+2,K+3
            a.x = wAs[col * K1 + kg];
            a.y = wAs[col * K1 + kg + 1];
            // B: 4x16 f32 — same K split across lane halves, N = col
            int ic0 = kl / 3,        t0 = kl - ic0 * 3;
            int ic1 = (kl + 1) / 3,  t1 = (kl + 1) - ic1 * 3;
            bm.x = xs[ic0 * XS_STR + nbase + t0];
            bm.y = xs[ic1 * XS_STR + nbase + t1];
            acc = __builtin_amdgcn_wmma_f32_16x16x4_f32(
                      false, a, false, bm, (short)0, acc, false, false);
        }
    }

    // bias + relu, C layout: VGPR r -> M=r (lanes 0-15) / M=r+8 (lanes 16-31)
    #pragma unroll
    for (int r = 0; r < 8; ++r) {
        int m = r + half * 8;
        float v = acc[r] + b1s[m];
        h1s[m * H_TILE + nbase] = fmaxf(v, 0.0f);
    }
    __syncthreads();

    // ---- conv2 (10x10, 1x1) + relu ; zero out-of-range h positions ----
    for (int i = tid; i < OC1 * H_TILE; i += 256) {
        int oc = i / H_TILE, p = i - oc * H_TILE;
        float a2 = b2s[oc];
        #pragma unroll
        for (int ic = 0; ic < OC1; ++ic)
            a2 += w2s[oc * 10 + ic] * h1s[ic * H_TILE + p];
        int t = hb + p;
        float v = fmaxf(a2, 0.0f);
        if (t < 0 || t >= NH) v = 0.0f;     // halo / padding for convT
        h2s[i] = v;
    }
    __syncthreads();

    // ---- conv-transpose (k=3) + relu, positions n = 0..125 ----
    for (int i = tid; i < OC1 * H_TILE; i += 256) {
        int oc = i / H_TILE, n = i - oc * H_TILE;
        if (n < 126) {
            float a3 = bts[oc];
            #pragma unroll
            for (int ic = 0; ic < OC1; ++ic) {
                const float* wrow = &wtA[oc * 30 + ic * 3];
                const float* hrow = &h2s[ic * H_TILE + n];
                a3 += wrow[0] * hrow[0];
                a3 += wrow[1] * hrow[1];
                a3 += wrow[2] * hrow[2];
            }
            ys[n * OC1 + oc] = fmaxf(a3, 0.0f);
        }
    }
    __syncthreads();

    // ---- conv3 (2x10, 1x1) + sigmoid -> left/right ----
    if (tid < 126) {
        float aL = b3s[0], aR = b3s[1];
        #pragma unroll
        for (int ic = 0; ic < OC1; ++ic) {
            float v = ys[tid * OC1 + ic];
            aL += w3s[ic]      * v;
            aR += w3s[10 + ic] * v;
        }
        lefts[tid]  = 1.0f / (1.0f + expf(-aL));
        rights[tid] = 1.0f / (1.0f + expf(-aR));
    }
    __syncthreads();

    // ---- mass stencil + log ; accumulate block partial sum ----
    float val = 0.0f;
    const int i = o0 + tid;
    if (tid < OUT_TILE && i < NOUT) {
        const float cst = cstp[0];
        const int cb = b * (NSIG - 1);          // curr_diag row stride 8191
        float mi = curr[cb + i + 1] * rights[tid + 1]
                 + curr[cb + i]     * lefts[tid + 1];
        float mo = rights[tid] + lefts[tid + 2];
        val = logf(cst * mi / mo);
        out[b * NOUT + i] = val;
    }
    red[tid] = val;
    __syncthreads();
    #pragma unroll
    for (int s = 128; s >= 1; s >>= 1) {
        if (tid < s) red[tid] += red[tid + s];
        __syncthreads();
    }
    if (tid == 0) partials[blockIdx.x] = red[0];
}

// Deterministic global-mean subtraction: every block re-reduces the partial
// sums (cheap), then grid-strides over the output.
__global__ __launch_bounds__(256)
void dlem_finalize(const float* __restrict__ partials, int nPartials,
                   float* __restrict__ out, int total)
{
    __shared__ float red[256];
    const int tid = threadIdx.x;
    float s = 0.0f;
    for (int i = tid; i < nPartials; i += 256) s += partials[i];
    red[tid] = s;
    __syncthreads();
    #pragma unroll
    for (int st = 128; st >= 1; st >>= 1) {
        if (tid < st) red[tid] += red[tid + st];
        __syncthreads();
    }
    const float mean = red[0] / (float)total;
    for (int i = blockIdx.x * 256 + tid; i < total; i += gridDim.x * 256)
        out[i] -= mean;
}

extern "C" void kernel_launch(void* const* d_in, const int* in_sizes, int n_in,
                              void* d_out, int out_size, void* d_ws, size_t ws_size,
                              hipStream_t stream) {
    const float* signal = (const float*)d_in[0];
    const float* curr   = (const float*)d_in[1];
    // d_in[2] = index_diag (int, ==1, baked into the stencil)
    const float* w1 = (const float*)d_in[3];
    const float* b1 = (const float*)d_in[4];
    const float* w2 = (const float*)d_in[5];
    const float* b2 = (const float*)d_in[6];
    const float* wt = (const float*)d_in[7];
    const float* bt = (const float*)d_in[8];
    const float* w3 = (const float*)d_in[9];
    const float* b3 = (const float*)d_in[10];
    const float* cst = (const float*)d_in[11];

    float* out      = (float*)d_out;
    float* partials = (float*)d_ws;     // BATCH*NTILES floats

    const int nblocks = BATCH * NTILES;  // 2144
    dlem_pipeline<<<dim3(nblocks), dim3(256), 0, stream>>>(
        signal, curr, w1, b1, w2, b2, wt, bt, w3, b3, cst, out, partials);
    dlem_finalize<<<dim3(256), dim3(256), 0, stream>>>(
        partials, nblocks, out, BATCH * NOUT);
}